// EncoderBlock_70549132804729
// MI455X (gfx1250) — compile-verified
//
#include <hip/hip_runtime.h>
#include <math.h>

#define Bz 32
#define Dz 128
#define Lz 1024
#define Hz 8
#define Kz 7
#define DLz (Dz * Lz)           // 131072
#define NEG_BIG (-1.0e30f)

typedef __attribute__((ext_vector_type(2))) float v2f;
typedef __attribute__((ext_vector_type(8))) float v8f;

__device__ __forceinline__ v8f wmma_f32(v2f a, v2f b, v8f c) {
  // V_WMMA_F32_16X16X4_F32 : D = A(16x4) * B(4x16) + C(16x16), fp32 throughout
  return __builtin_amdgcn_wmma_f32_16x16x4_f32(false, a, false, b, (short)0, c,
                                               false, false);
}

// ---------------------------------------------------------------------------
// Fused residual + 2D LayerNorm over (D,L) per batch.
// mode 0: pre = a + b        (b broadcast over batch: pos_enc)
// mode 1: pre = relu(a) + b  (b per-batch residual)
// mode 2: pre = a + b        (b per-batch residual)
// Writes res = pre and out = LN(pre)*w + bias. b/res may alias (read-then-
// write same element per thread), so those params are NOT __restrict__.
// ---------------------------------------------------------------------------
__global__ void k_res_ln(const float* __restrict__ a, const float* b,
                         const float* __restrict__ w,
                         const float* __restrict__ bias,
                         float* res, float* __restrict__ out, int mode) {
  const int bb = blockIdx.x;
  const int tid = threadIdx.x;  // 512 threads
  const float* ap = a + (size_t)bb * DLz;
  const float* bp = (mode == 0) ? b : (b + (size_t)bb * DLz);
  float* rp = res + (size_t)bb * DLz;
  float* op = out + (size_t)bb * DLz;

  __shared__ double sh1[512];
  __shared__ double sh2[512];
  __shared__ float mu_s, rstd_s;

  double s1 = 0.0, s2 = 0.0;
  for (int i = tid; i < DLz; i += 512) {
    float av = ap[i];
    float pre = (mode == 1) ? (fmaxf(av, 0.0f) + bp[i]) : (av + bp[i]);
    rp[i] = pre;
    s1 += (double)pre;
    s2 += (double)pre * (double)pre;
  }
  sh1[tid] = s1;
  sh2[tid] = s2;
  __syncthreads();
  for (int off = 256; off > 0; off >>= 1) {
    if (tid < off) { sh1[tid] += sh1[tid + off]; sh2[tid] += sh2[tid + off]; }
    __syncthreads();
  }
  if (tid == 0) {
    double mu = sh1[0] / (double)DLz;
    double var = sh2[0] / (double)DLz - mu * mu;
    mu_s = (float)mu;
    rstd_s = (float)(1.0 / sqrt(var + 1e-5));
  }
  __syncthreads();
  float mu = mu_s, rstd = rstd_s;
  for (int i = tid; i < DLz; i += 512) {
    float pre = rp[i];
    op[i] = (pre - mu) * rstd * w[i] + bias[i];
  }
}

// ---------------------------------------------------------------------------
// Depthwise conv, K=7, zero pad 3. x,y: (B,D,L). w: (D,7), wb: (D).
// ---------------------------------------------------------------------------
__global__ void k_dwconv(const float* __restrict__ x,
                         const float* __restrict__ w,
                         const float* __restrict__ wb,
                         float* __restrict__ y) {
  int idx = blockIdx.x * 256 + threadIdx.x;  // over B*D*L
  int l = idx & (Lz - 1);
  int c = (idx >> 10) & (Dz - 1);
  const float* xr = x + (idx - l);
  float acc = wb[c];
#pragma unroll
  for (int k = 0; k < Kz; k++) {
    int ll = l + k - 3;
    if (ll >= 0 && ll < Lz) acc = fmaf(w[c * Kz + k], xr[ll], acc);
  }
  y[idx] = acc;
}

// ---------------------------------------------------------------------------
// WMMA GEMM: Y[b,o,l] = sum_c W[o,c] * X[b,c,l] + bias[o]
// grid = B*64 blocks, 256 threads (8 waves). Wave = one 16(M=o) x 16(N=l)
// tile, K loop over D=128 in steps of 4 (32 chained f32 WMMAs).
// ---------------------------------------------------------------------------
__global__ void k_gemm(const float* __restrict__ W,
                       const float* __restrict__ bias,
                       const float* __restrict__ X, float* __restrict__ Y) {
  const int wave = threadIdx.x >> 5;
  const int lane = threadIdx.x & 31;
  const int lm = lane & 15;
  const int hi = lane >> 4;
  const int nt = blockIdx.x & 63;
  const int b = blockIdx.x >> 6;
  const int m0 = wave * 16;
  const int l0 = nt * 16;
  const float* Xb = X + (size_t)b * DLz;
  float* Yb = Y + (size_t)b * DLz;

  v8f c = {};
#pragma unroll 4
  for (int k0 = 0; k0 < Dz; k0 += 4) {
    // A tile (16x4 of W): lane m holds (M=m0+lm, K=k0+2*hi+{0,1})
    const float* wp = W + (size_t)(m0 + lm) * Dz + k0 + 2 * hi;
    v2f a;
    a.x = wp[0];
    a.y = wp[1];
    // B tile (4x16 of X): VGPR0 rows {k0, k0+2}, VGPR1 rows {k0+1, k0+3}
    const float* xp = Xb + (size_t)(k0 + 2 * hi) * Lz + l0 + lm;
    v2f bv;
    bv.x = xp[0];
    bv.y = xp[Lz];
    c = wmma_f32(a, bv, c);
  }
#pragma unroll
  for (int j = 0; j < 8; j++) {
    int o = m0 + j + 8 * hi;  // C/D: lane half hi holds rows j / j+8
    Yb[(size_t)o * Lz + l0 + lm] = c[j] + bias[o];
  }
}

// ---------------------------------------------------------------------------
// Flash attention. Q,K,V,O in (B, D, L) with head h owning channels
// [16h,16h+16). One wave per (b, h, 16-query tile); online softmax over 64
// key tiles; QK^T and P*V both via f32 WMMA. P needs C->A layout transpose:
// bounce through a per-wave LDS tile.
// ---------------------------------------------------------------------------
__global__ void k_attn(const float* __restrict__ Q, const float* __restrict__ K,
                       const float* __restrict__ V,
                       const float* __restrict__ mask,
                       float* __restrict__ O) {
  __shared__ float plds[8][16][17];
  const int wave = threadIdx.x >> 5;
  const int lane = threadIdx.x & 31;
  const int lm = lane & 15;
  const int hi = lane >> 4;
  const int task = blockIdx.x * 8 + wave;  // B*H*64 = 16384 tasks
  const int qt = task & 63;
  const int h = (task >> 6) & 7;
  const int b = task >> 9;
  const int q0 = qt * 16;
  const size_t base = ((size_t)b * Dz + h * 16) * Lz;
  const float* Qb = Q + base;
  const float* Kb = K + base;
  const float* Vb = V + base;
  const float* mb = mask + (size_t)b * Lz;

  // Q tile in A layout: A[M=qi][K=d] = Qb[d*L + q0+qi]
  v2f aQ[4];
#pragma unroll
  for (int s = 0; s < 4; s++) {
    int d = 4 * s + 2 * hi;
    aQ[s].x = Qb[(size_t)d * Lz + q0 + lm];
    aQ[s].y = Qb[(size_t)(d + 1) * Lz + q0 + lm];
  }

  v8f oacc = {};
  float rowm[8], rowl[8];
#pragma unroll
  for (int j = 0; j < 8; j++) { rowm[j] = -3.0e38f; rowl[j] = 0.0f; }
  const float scale = 0.25f;  // 1/sqrt(16)

  for (int kt = 0; kt < Lz / 16; kt++) {
    const int kp0 = kt * 16;
    // S = Q * K^T : B[K=d][N=ki] = Kb[d*L + kp0+ki]
    v8f sc = {};
#pragma unroll
    for (int s = 0; s < 4; s++) {
      int d = 4 * s + 2 * hi;
      v2f bv;
      bv.x = Kb[(size_t)d * Lz + kp0 + lm];
      bv.y = Kb[(size_t)(d + 1) * Lz + kp0 + lm];
      sc = wmma_f32(aQ[s], bv, sc);
    }
    const float mval = mb[kp0 + lm];  // key mask for this lane's column
    float p[8];
#pragma unroll
    for (int j = 0; j < 8; j++) {
      float s = sc[j] * scale;
      s = (mval > 0.5f) ? NEG_BIG : s;
      float m = s;  // row max across the 16 lanes holding this row
      for (int off = 8; off > 0; off >>= 1)
        m = fmaxf(m, __shfl_xor(m, off, 16));
      float nm = fmaxf(rowm[j], m);
      float pe = __expf(s - nm);
      float ps = pe;
      for (int off = 8; off > 0; off >>= 1) ps += __shfl_xor(ps, off, 16);
      float corr = __expf(rowm[j] - nm);
      rowl[j] = rowl[j] * corr + ps;
      rowm[j] = nm;
      oacc[j] *= corr;
      p[j] = pe;
    }
    // C-layout -> A-layout for P via per-wave LDS tile
#pragma unroll
    for (int j = 0; j < 8; j++) plds[wave][j + 8 * hi][lm] = p[j];
    asm volatile("s_wait_dscnt 0" ::: "memory");
#pragma unroll
    for (int s = 0; s < 4; s++) {
      int kk = 4 * s + 2 * hi;
      v2f ap;
      ap.x = plds[wave][lm][kk];
      ap.y = plds[wave][lm][kk + 1];
      // B[K=ki][N=d] = Vb[d*L + kp0+ki] -> contiguous float pair per lane
      const float* vp = Vb + (size_t)lm * Lz + kp0 + kk;
      v2f bv;
      bv.x = vp[0];
      bv.y = vp[1];
      oacc = wmma_f32(ap, bv, oacc);
    }
  }
  // O[b, 16h+d, q0+qi] = oacc / rowl  (row qi = j + 8*hi, col d = lm)
  float* Ob = O + base;
#pragma unroll
  for (int j = 0; j < 8; j++)
    Ob[(size_t)lm * Lz + q0 + j + 8 * hi] = oacc[j] / rowl[j];
}

// ---------------------------------------------------------------------------
// Final: out = relu(g) + res
// ---------------------------------------------------------------------------
__global__ void k_relu_add(const float* __restrict__ g,
                           const float* __restrict__ r,
                           float* __restrict__ y) {
  int i = blockIdx.x * 256 + threadIdx.x;
  y[i] = fmaxf(g[i], 0.0f) + r[i];
}

// ---------------------------------------------------------------------------
extern "C" void kernel_launch(void* const* d_in, const int* in_sizes, int n_in,
                              void* d_out, int out_size, void* d_ws,
                              size_t ws_size, hipStream_t stream) {
  (void)in_sizes; (void)n_in; (void)out_size; (void)ws_size;
  const float* x       = (const float*)d_in[0];
  const float* mask    = (const float*)d_in[1];
  const float* pe      = (const float*)d_in[2];
  const float* normb_w = (const float*)d_in[3];
  const float* normb_b = (const float*)d_in[4];
  const float* dw_w    = (const float*)d_in[5];
  const float* dw_b    = (const float*)d_in[6];
  const float* pw_w    = (const float*)d_in[7];
  const float* pw_b    = (const float*)d_in[8];
  const float* norms_w = (const float*)d_in[9];
  const float* norms_b = (const float*)d_in[10];
  const float* qw      = (const float*)d_in[11];
  const float* qb      = (const float*)d_in[12];
  const float* kw      = (const float*)d_in[13];
  const float* kb      = (const float*)d_in[14];
  const float* vw      = (const float*)d_in[15];
  const float* vb      = (const float*)d_in[16];
  const float* aw      = (const float*)d_in[17];
  const float* ab      = (const float*)d_in[18];
  const float* norme_w = (const float*)d_in[19];
  const float* norme_b = (const float*)d_in[20];
  const float* fw      = (const float*)d_in[21];
  const float* fb      = (const float*)d_in[22];
  float* out_f = (float*)d_out;

  const size_t S = (size_t)Bz * Dz * Lz;  // 4,194,304 floats per slab
  float* ws = (float*)d_ws;
  float* res = ws + 0 * S;
  float* cur = ws + 1 * S;
  float* tA  = ws + 2 * S;
  float* tB  = ws + 3 * S;
  float* tC  = ws + 4 * S;
  float* tD  = ws + 5 * S;

  const int ELEM_BLKS = (int)(S / 256);      // 16384
  const int GEMM_BLKS = Bz * (Lz / 16);      // 2048
  const int ATTN_BLKS = Bz * Hz * (Lz / 16) / 8;  // 2048

  // x + pos_enc -> res ; LayerNorm -> cur
  k_res_ln<<<Bz, 512, 0, stream>>>(x, pe, normb_w, normb_b, res, cur, 0);

  // 4 conv blocks
  for (int i = 0; i < 4; i++) {
    k_dwconv<<<ELEM_BLKS, 256, 0, stream>>>(cur, dw_w + i * Dz * Kz,
                                            dw_b + i * Dz, tA);
    k_gemm<<<GEMM_BLKS, 256, 0, stream>>>(pw_w + (size_t)i * Dz * Dz,
                                          pw_b + i * Dz, tA, tB);
    k_res_ln<<<Bz, 512, 0, stream>>>(tB, res, norms_w + (size_t)i * DLz,
                                     norms_b + (size_t)i * DLz, res, cur, 1);
  }

  // attention: Q/K/V projections, flash attention, output projection
  k_gemm<<<GEMM_BLKS, 256, 0, stream>>>(qw, qb, cur, tA);
  k_gemm<<<GEMM_BLKS, 256, 0, stream>>>(kw, kb, cur, tB);
  k_gemm<<<GEMM_BLKS, 256, 0, stream>>>(vw, vb, cur, tC);
  k_attn<<<ATTN_BLKS, 256, 0, stream>>>(tA, tB, tC, mask, tD);
  k_gemm<<<GEMM_BLKS, 256, 0, stream>>>(aw, ab, tD, tA);
  k_res_ln<<<Bz, 512, 0, stream>>>(tA, res, norme_w, norme_b, res, cur, 2);

  // feed-forward + final relu-residual
  k_gemm<<<GEMM_BLKS, 256, 0, stream>>>(fw, fb, cur, tB);
  k_relu_add<<<ELEM_BLKS, 256, 0, stream>>>(tB, res, out_f);
}